// Adj_68856915689818
// MI455X (gfx1250) — compile-verified
//
#include <hip/hip_runtime.h>
#include <hip/hip_bf16.h>

typedef __attribute__((ext_vector_type(2))) float v2f;
typedef __attribute__((ext_vector_type(4))) float v4f;
typedef __attribute__((ext_vector_type(8))) float v8f;

#define BATCH 2
#define NPTS  1024
#define CDIM  64
#define TILE  64
#define LDSS  68   // padded LDS row stride (floats): bank = 4*row + col, conflict-free

// ---------------------------------------------------------------------------
// Kernel 1: Y = X @ W   (y_i = W^T x_i as a row), and r_i = ||y_i||^2
// One 64-thread block per row of X (B*N rows total).
// ---------------------------------------------------------------------------
__global__ __launch_bounds__(64) void proj_kernel(const float* __restrict__ X,
                                                  const float* __restrict__ W,
                                                  float* __restrict__ Y,
                                                  float* __restrict__ r) {
    __shared__ __align__(16) float xrow[CDIM];
    __shared__ __align__(16) float red[CDIM];
    const int row = blockIdx.x;      // 0 .. B*N-1
    const int t   = threadIdx.x;     // 0 .. 63
    xrow[t] = X[row * CDIM + t];
    __syncthreads();
    float acc = 0.f;
#pragma unroll 16
    for (int k = 0; k < CDIM; ++k) {
        acc = fmaf(xrow[k], W[k * CDIM + t], acc);   // column t of W, coalesced
    }
    Y[row * CDIM + t] = acc;
    red[t] = acc * acc;
    __syncthreads();
#pragma unroll
    for (int s = 32; s > 0; s >>= 1) {
        if (t < s) red[t] += red[t + s];
        __syncthreads();
    }
    if (t == 0) r[row] = red[0];
}

// ---------------------------------------------------------------------------
// Kernel 2: 64x64 tile of G = Y Y^T via V_WMMA_F32_16X16X4_F32, fused epilogue
//   adj[b,i,j] = exp(-(r_i + r_j - 2*G_ij)) + (i==j)
// Block = 128 threads = 4 wave32; wave w computes M-strip rows [w*16, w*16+16).
// ---------------------------------------------------------------------------
__global__ __launch_bounds__(128) void adj_kernel(const float* __restrict__ Y,
                                                  const float* __restrict__ r,
                                                  float* __restrict__ out) {
    __shared__ __align__(16) float As[TILE * LDSS];   // rows ti..ti+63 of Y
    __shared__ __align__(16) float Bs[TILE * LDSS];   // rows tj..tj+63 of Y
    const int b   = blockIdx.z;
    const int ti  = blockIdx.y * TILE;
    const int tj  = blockIdx.x * TILE;
    const int tid = threadIdx.x;

    const float* Yb = Y + (size_t)b * NPTS * CDIM;
    const float* rb = r + (size_t)b * NPTS;

    // Cooperative load: 16 threads cover one 64-float row as 16 float4s;
    // 128 threads -> 8 rows per pass, 8 passes for 64 rows (A and B tiles).
    {
        const int cg = (tid & 15) * 4;   // column (float4 granules)
        const int r0 = tid >> 4;         // 0..7
#pragma unroll
        for (int p = 0; p < 8; ++p) {
            const int rr = r0 + p * 8;
            v4f va = *(const v4f*)(Yb + (size_t)(ti + rr) * CDIM + cg);
            v4f vb = *(const v4f*)(Yb + (size_t)(tj + rr) * CDIM + cg);
            *(v4f*)(&As[rr * LDSS + cg]) = va;
            *(v4f*)(&Bs[rr * LDSS + cg]) = vb;
        }
    }
    __syncthreads();

    const int wave = tid >> 5;       // 0..3 -> which 16-row M strip
    const int lane = tid & 31;
    const int l16  = lane & 15;
    const int half = lane >> 4;      // ISA 16x4 f32 A layout: half selects K pair

    v8f acc0 = {}, acc1 = {}, acc2 = {}, acc3 = {};

    const float* Arow = &As[(wave * 16 + l16) * LDSS];
    const float* B0   = &Bs[( 0 + l16) * LDSS];
    const float* B1   = &Bs[(16 + l16) * LDSS];
    const float* B2   = &Bs[(32 + l16) * LDSS];
    const float* B3   = &Bs[(48 + l16) * LDSS];

#pragma unroll
    for (int kk = 0; kk < CDIM; kk += 4) {
        const int k = kk + 2 * half;   // lanes 0-15: K=kk,kk+1 ; lanes 16-31: K=kk+2,kk+3
        v2f a;  a.x = Arow[k]; a.y = Arow[k + 1];
        v2f bv;
        bv.x = B0[k]; bv.y = B0[k + 1];
        acc0 = __builtin_amdgcn_wmma_f32_16x16x4_f32(false, a, false, bv, (short)0, acc0, false, false);
        bv.x = B1[k]; bv.y = B1[k + 1];
        acc1 = __builtin_amdgcn_wmma_f32_16x16x4_f32(false, a, false, bv, (short)0, acc1, false, false);
        bv.x = B2[k]; bv.y = B2[k + 1];
        acc2 = __builtin_amdgcn_wmma_f32_16x16x4_f32(false, a, false, bv, (short)0, acc2, false, false);
        bv.x = B3[k]; bv.y = B3[k + 1];
        acc3 = __builtin_amdgcn_wmma_f32_16x16x4_f32(false, a, false, bv, (short)0, acc3, false, false);
    }

    // Epilogue: C/D layout -> row i = v + 8*half within strip, col j = l16.
    float* outb = out + (size_t)b * NPTS * NPTS;
    const int i_base = ti + wave * 16 + half * 8;
    const v8f accs[4] = {acc0, acc1, acc2, acc3};
#pragma unroll
    for (int ns = 0; ns < 4; ++ns) {
        const int j  = tj + ns * 16 + l16;
        const float rj = rb[j];
#pragma unroll
        for (int v = 0; v < 8; ++v) {
            const int i = i_base + v;
            const float g = accs[ns][v];
            const float q = rb[i] + rj - 2.0f * g;
            outb[(size_t)i * NPTS + j] = __expf(-q) + (i == j ? 1.0f : 0.0f);
        }
    }
}

extern "C" void kernel_launch(void* const* d_in, const int* in_sizes, int n_in,
                              void* d_out, int out_size, void* d_ws, size_t ws_size,
                              hipStream_t stream) {
    const float* X = (const float*)d_in[0];   // (B, N, C) f32
    const float* W = (const float*)d_in[1];   // (C, C) f32
    float* out = (float*)d_out;               // (B, N, N) f32

    float* Yw = (float*)d_ws;                      // B*N*C floats
    float* rw = Yw + (size_t)BATCH * NPTS * CDIM;  // B*N floats

    proj_kernel<<<BATCH * NPTS, 64, 0, stream>>>(X, W, Yw, rw);

    dim3 grid(NPTS / TILE, NPTS / TILE, BATCH);   // 16 x 16 x 2 = 512 workgroups
    adj_kernel<<<grid, 128, 0, stream>>>(Yw, rw, out);
}